// SuperGATConvNN_86938728005822
// MI455X (gfx1250) — compile-verified
//
#include <hip/hip_runtime.h>
#include <hip/hip_bf16.h>
#include <math.h>

#define HEADS 4
#define HID 32
#define CH 128  // HEADS*HID

typedef float v2f __attribute__((ext_vector_type(2)));
typedef float v8f __attribute__((ext_vector_type(8)));
typedef int   v4i __attribute__((vector_size(4 * sizeof(int))));  // matches builtin param

#if defined(__has_builtin)
#if __has_builtin(__builtin_amdgcn_global_load_async_to_lds_b128)
#define HAVE_ASYNC_LDS 1
#endif
#endif

// ---------------------------------------------------------------------------
// Ordered-uint encoding of float so unsigned atomicMax == float max.
// ---------------------------------------------------------------------------
__device__ __forceinline__ unsigned ord_encode(float f) {
    unsigned u = __float_as_uint(f);
    return (u & 0x80000000u) ? ~u : (u | 0x80000000u);
}
__device__ __forceinline__ float ord_decode(unsigned u) {
    unsigned b = (u & 0x80000000u) ? (u & 0x7FFFFFFFu) : ~u;
    return __uint_as_float(b);
}

// ---------------------------------------------------------------------------
// f32 GEMM with V_WMMA_F32_16X16X4_F32, B staged in LDS (async global->LDS).
// Compile-time K/NC so all LDS/global offsets fold into instruction immediates.
// Block = one 16-row stripe of C; wave w = 16-col tile w; blockDim = (NC/16)*32.
// Static LDS = K*NC floats (64KB for NC=128, 32KB for NC=64).
// A layout (ISA 7.12.2): lanes 0-15 rows M, VGPR0/1 = K {0,1}; lanes 16-31 K {2,3}.
// B layout (mirror):     lanes 0-15 cols N, VGPR0/1 = K {0,1}; lanes 16-31 K {2,3}.
// D layout: VGPR r, lanes 0-15 -> M=r, lanes 16-31 -> M=8+r, N=lane&15.
// ---------------------------------------------------------------------------
template <int K, int NC>
__global__ void wmma_gemm_f32_lds(const float* __restrict__ A,
                                  const float* __restrict__ B,
                                  const float* __restrict__ bias,
                                  float* __restrict__ C) {
    __shared__ float sB[K * NC];
    constexpr int NTHREADS = (NC / 16) * 32;
    const int tid = threadIdx.x;

    // ---- stage B panel into LDS via gfx1250 async global->LDS when available
#if HAVE_ASYNC_LDS
#pragma unroll
    for (int i = tid * 4; i < K * NC; i += NTHREADS * 4) {
        __builtin_amdgcn_global_load_async_to_lds_b128(
            (__attribute__((address_space(1))) v4i*)(B + i),
            (__attribute__((address_space(3))) v4i*)(sB + i),
            /*offset=*/0, /*cpol=*/0);
    }
#if __has_builtin(__builtin_amdgcn_s_wait_asynccnt)
    __builtin_amdgcn_s_wait_asynccnt(0);
#else
    asm volatile("s_wait_asynccnt 0x0" ::: "memory");
#endif
#else
#pragma unroll
    for (int i = tid * 4; i < K * NC; i += NTHREADS * 4)
        *(float4*)&sB[i] = *(const float4*)&B[i];
#endif
    __syncthreads();

    const int wave = tid >> 5;     // column tile
    const int lane = tid & 31;
    const int half = lane >> 4;    // 0: K offset {0,1}; 1: K offset {2,3}
    const int l15  = lane & 15;
    const int ak   = half * 2;
    const int trow = blockIdx.x;

    const float* arow = A + (size_t)(trow * 16 + l15) * K + ak;
    const float* brow = sB + wave * 16 + l15 + ak * NC;

    v8f acc = {};
#pragma unroll
    for (int k = 0; k < K; k += 4) {
        v2f a, b;
        a.x = arow[k];                 // A[row][k+ak]
        a.y = arow[k + 1];             // A[row][k+ak+1]
        b.x = brow[k * NC];            // B[k+ak][col]      (imm LDS offsets)
        b.y = brow[k * NC + NC];       // B[k+ak+1][col]
        acc = __builtin_amdgcn_wmma_f32_16x16x4_f32(
            /*neg_a=*/false, a, /*neg_b=*/false, b,
            /*c_mod=*/(short)0, acc, /*reuse_a=*/false, /*reuse_b=*/false);
    }

    float bv = bias ? bias[wave * 16 + l15] : 0.f;
    float* crow = C + (size_t)(trow * 16 + half * 8) * NC + wave * 16 + l15;
#pragma unroll
    for (int r = 0; r < 8; ++r)
        crow[(size_t)r * NC] = acc[r] + bv;
}

// ---------------------------------------------------------------------------
// hl[n,h] = <h[n,h,:], att_l[h,:]>,  hr likewise.
// ---------------------------------------------------------------------------
__global__ void node_att_kernel(const float* __restrict__ h,
                                const float* __restrict__ att_l,
                                const float* __restrict__ att_r,
                                float* __restrict__ hl, float* __restrict__ hr,
                                int n) {
    int t = blockIdx.x * blockDim.x + threadIdx.x;
    if (t >= n * HEADS) return;
    int node = t >> 2, hd = t & 3;
    const float* row = h + (size_t)node * CH + hd * HID;
    const float* al = att_l + hd * HID;
    const float* ar = att_r + hd * HID;
    float sl = 0.f, sr = 0.f;
#pragma unroll
    for (int c = 0; c < HID; ++c) { float v = row[c]; sl += v * al[c]; sr += v * ar[c]; }
    hl[t] = sl; hr[t] = sr;
}

// ---------------------------------------------------------------------------
// Pass A: alpha = leaky_relu((hl[src]+hr[dst]) * sigmoid(<h_src,h_dst>)),
// atomic segment-max over dst via ordered-uint max.
// Edges e<E come from edge_index; e>=E are self-loops (src=dst=e-E).
// ---------------------------------------------------------------------------
__global__ void edge_alpha_kernel(const int* __restrict__ ei,
                                  const float* __restrict__ h,
                                  const float* __restrict__ hl,
                                  const float* __restrict__ hr,
                                  float* __restrict__ alpha,
                                  unsigned* __restrict__ amax,
                                  int E, int n) {
    int t = blockIdx.x * blockDim.x + threadIdx.x;
    int etot = E + n;
    if (t >= etot * HEADS) return;
    int e = t >> 2, hd = t & 3;
    int src = (e < E) ? ei[e]     : (e - E);
    int dst = (e < E) ? ei[E + e] : (e - E);

    const float4* ps = (const float4*)(h + (size_t)src * CH + hd * HID);
    const float4* pd = (const float4*)(h + (size_t)dst * CH + hd * HID);
    float dot = 0.f;
#pragma unroll
    for (int i = 0; i < HID / 4; ++i) {
        float4 a = ps[i], b = pd[i];
        dot += a.x * b.x + a.y * b.y + a.z * b.z + a.w * b.w;
    }
    float al = hl[src * HEADS + hd] + hr[dst * HEADS + hd];
    al = al / (1.f + expf(-dot));                 // * sigmoid(logits)
    al = (al > 0.f) ? al : 0.2f * al;             // leaky_relu(0.2)
    alpha[t] = al;
    atomicMax(&amax[dst * HEADS + hd], ord_encode(al));
}

// ---------------------------------------------------------------------------
// Pass B: ex = exp(alpha - amax[dst]); alpha <- ex; denom[dst] += ex.
// ---------------------------------------------------------------------------
__global__ void edge_exp_kernel(const int* __restrict__ ei,
                                float* __restrict__ alpha,
                                const unsigned* __restrict__ amax,
                                float* __restrict__ denom,
                                int E, int n) {
    int t = blockIdx.x * blockDim.x + threadIdx.x;
    int etot = E + n;
    if (t >= etot * HEADS) return;
    int e = t >> 2, hd = t & 3;
    int dst = (e < E) ? ei[E + e] : (e - E);
    float m = ord_decode(amax[dst * HEADS + hd]);
    float ex = expf(alpha[t] - m);
    alpha[t] = ex;
    atomicAdd(&denom[dst * HEADS + hd], ex);
}

// ---------------------------------------------------------------------------
// Pass C: one thread per (edge, channel): fully coalesced gather + atomics.
// accum[dst,c] += h[src,c] * (ex[e,hd] / (denom[dst,hd] + 1e-16)).
// ---------------------------------------------------------------------------
__global__ void edge_scatter_kernel(const int* __restrict__ ei,
                                    const float* __restrict__ h,
                                    const float* __restrict__ alpha,
                                    const float* __restrict__ denom,
                                    float* __restrict__ accum,
                                    int E, int n) {
    int t = blockIdx.x * blockDim.x + threadIdx.x;
    int etot = E + n;
    if (t >= etot * CH) return;         // 108.8M threads, fits int32
    int e = t >> 7;                     // CH == 128
    int c = t & (CH - 1);
    int hd = c >> 5;                    // HID == 32
    int src = (e < E) ? ei[e]     : (e - E);
    int dst = (e < E) ? ei[E + e] : (e - E);
    float a = alpha[e * HEADS + hd] / (denom[dst * HEADS + hd] + 1e-16f);
    atomicAdd(&accum[(size_t)dst * CH + c], h[(size_t)src * CH + c] * a);
}

// ---------------------------------------------------------------------------
// accum <- elu(accum + bias)  (in place)
// ---------------------------------------------------------------------------
__global__ void elu_bias_kernel(float* __restrict__ accum,
                                const float* __restrict__ bias, int n) {
    int t = blockIdx.x * blockDim.x + threadIdx.x;
    if (t >= n * CH) return;
    int c = t & (CH - 1);
    float v = accum[t] + bias[c];
    accum[t] = (v > 0.f) ? v : (expf(v) - 1.f);
}

// ---------------------------------------------------------------------------
extern "C" void kernel_launch(void* const* d_in, const int* in_sizes, int n_in,
                              void* d_out, int out_size, void* d_ws, size_t ws_size,
                              hipStream_t stream) {
    const float* x     = (const float*)d_in[0];
    const int*   ei    = (const int*)d_in[1];    // jax default x64-off => int32
    const float* W     = (const float*)d_in[2];
    const float* att_l = (const float*)d_in[3];
    const float* att_r = (const float*)d_in[4];
    const float* bias  = (const float*)d_in[5];
    const float* W2    = (const float*)d_in[6];
    const float* b2    = (const float*)d_in[7];
    float* out = (float*)d_out;

    const int n    = in_sizes[0] / CH;   // 50000
    const int E    = in_sizes[1] / 2;    // 800000
    const int etot = E + n;

    // workspace layout (all 16B-aligned: sizes are multiples of 16 bytes)
    char* p = (char*)d_ws;
    float*    h     = (float*)p;    p += (size_t)n * CH * sizeof(float);
    float*    accum = (float*)p;    p += (size_t)n * CH * sizeof(float);
    float*    alpha = (float*)p;    p += (size_t)etot * HEADS * sizeof(float);
    float*    hl    = (float*)p;    p += (size_t)n * HEADS * sizeof(float);
    float*    hr    = (float*)p;    p += (size_t)n * HEADS * sizeof(float);
    unsigned* amax  = (unsigned*)p; p += (size_t)n * HEADS * sizeof(unsigned);
    float*    denom = (float*)p;    p += (size_t)n * HEADS * sizeof(float);

    (void)hipMemsetAsync(accum, 0, (size_t)n * CH * sizeof(float), stream);
    (void)hipMemsetAsync(amax,  0, (size_t)n * HEADS * sizeof(unsigned), stream); // 0 < every encoded float
    (void)hipMemsetAsync(denom, 0, (size_t)n * HEADS * sizeof(float), stream);

    // 1) h = x @ W   [n,128] = [n,128]@[128,128]; one block per 16-row stripe,
    //    8 waves = 8 column tiles, B (64KB) staged in LDS.
    wmma_gemm_f32_lds<CH, CH><<<n / 16, (CH / 16) * 32, 0, stream>>>(x, W, nullptr, h);

    // 2) attention projections
    {
        int cnt = n * HEADS;
        node_att_kernel<<<(cnt + 255) / 256, 256, 0, stream>>>(h, att_l, att_r, hl, hr, n);
    }
    // 3) edge passes
    {
        int cnt = etot * HEADS;
        int grid = (cnt + 255) / 256;
        edge_alpha_kernel<<<grid, 256, 0, stream>>>(ei, h, hl, hr, alpha, amax, E, n);
        edge_exp_kernel<<<grid, 256, 0, stream>>>(ei, alpha, amax, denom, E, n);
        int cnt2 = etot * CH;
        edge_scatter_kernel<<<(cnt2 + 255) / 256, 256, 0, stream>>>(ei, h, alpha, denom, accum, E, n);
    }
    // 4) bias + ELU in place
    {
        int cnt = n * CH;
        elu_bias_kernel<<<(cnt + 255) / 256, 256, 0, stream>>>(accum, bias, n);
    }
    // 5) out = act @ W2 + b2   [n,64]; 4 waves per block, B2 (32KB) in LDS.
    wmma_gemm_f32_lds<CH, 64><<<n / 16, (64 / 16) * 32, 0, stream>>>(accum, W2, b2, out);
}